// GNNLayer_1400159339040
// MI455X (gfx1250) — compile-verified
//
#include <hip/hip_runtime.h>

// N_NODES=50000, N_EDGES=800000, IN_DIM=64
#define DIM 64

typedef __attribute__((ext_vector_type(2))) float v2f;
typedef __attribute__((ext_vector_type(8))) float v8f;

__device__ __forceinline__ unsigned f2key(float f) {
    unsigned b = __float_as_uint(f);
    return (b & 0x80000000u) ? ~b : (b | 0x80000000u);
}
__device__ __forceinline__ float key2f(unsigned u) {
    unsigned b = (u & 0x80000000u) ? (u & 0x7fffffffu) : ~u;
    return __uint_as_float(b);
}

// ---------- 0: zero h_agg + stats (maxkey, sum) ----------
__global__ void zero_kernel(float* __restrict__ p, int n) {
    int i = blockIdx.x * blockDim.x + threadIdx.x;
    if (i < n) p[i] = 0.0f;
}

// ---------- 1: per-edge attention logits (wave per edge, coalesced row loads) ----------
__global__ void edge_logits_kernel(const float* __restrict__ h,
                                   const int* __restrict__ src,
                                   const int* __restrict__ dst,
                                   const float* __restrict__ W_att,
                                   const float* __restrict__ b_att,
                                   float* __restrict__ logits, int E) {
    int wave = (blockIdx.x * blockDim.x + threadIdx.x) >> 5;
    int lane = threadIdx.x & 31;
    if (wave >= E) return;
    long long s = src[wave], d = dst[wave];
    float p = h[s * DIM + lane]      * W_att[lane]
            + h[s * DIM + 32 + lane] * W_att[32 + lane]
            + h[d * DIM + lane]      * W_att[64 + lane]
            + h[d * DIM + 32 + lane] * W_att[96 + lane];
    #pragma unroll
    for (int off = 16; off > 0; off >>= 1) p += __shfl_xor(p, off, 32);
    if (lane == 0) {
        float x = p + b_att[0];
        logits[wave] = x > 0.0f ? x : 0.2f * x;   // leaky_relu(0.2)
    }
}

// ---------- 2: global max over logits ----------
__global__ void max_kernel(const float* __restrict__ logits, int E,
                           unsigned* __restrict__ maxkey) {
    __shared__ unsigned sm[256];
    unsigned k = 0u;  // decodes below -inf key; every real logit dominates it
    for (int i = blockIdx.x * blockDim.x + threadIdx.x; i < E;
         i += gridDim.x * blockDim.x) {
        unsigned kk = f2key(logits[i]);
        k = kk > k ? kk : k;
    }
    sm[threadIdx.x] = k;
    __syncthreads();
    for (int s = 128; s > 0; s >>= 1) {
        if (threadIdx.x < s) {
            unsigned o = sm[threadIdx.x + s];
            if (o > sm[threadIdx.x]) sm[threadIdx.x] = o;
        }
        __syncthreads();
    }
    if (threadIdx.x == 0) atomicMax(maxkey, sm[0]);
}

// ---------- 3: exp(l - max) in place + global sum ----------
__global__ void expsum_kernel(float* __restrict__ logits, int E,
                              const unsigned* __restrict__ maxkey,
                              float* __restrict__ sumf) {
    __shared__ float sm[256];
    float m = key2f(*maxkey);
    float acc = 0.0f;
    for (int i = blockIdx.x * blockDim.x + threadIdx.x; i < E;
         i += gridDim.x * blockDim.x) {
        float e = __expf(logits[i] - m);
        logits[i] = e;
        acc += e;
    }
    sm[threadIdx.x] = acc;
    __syncthreads();
    for (int s = 128; s > 0; s >>= 1) {
        if (threadIdx.x < s) sm[threadIdx.x] += sm[threadIdx.x + s];
        __syncthreads();
    }
    if (threadIdx.x == 0) unsafeAtomicAdd(sumf, sm[0]);
}

// ---------- 4: normalize attn, write attn out, scatter msg into h_agg ----------
__global__ void scatter_kernel(const float* __restrict__ h,
                               const int* __restrict__ src,
                               const int* __restrict__ dst,
                               const float* __restrict__ expv,
                               const float* __restrict__ sumf,
                               float* __restrict__ attn_out,
                               float* __restrict__ h_agg, int E) {
    int wave = (blockIdx.x * blockDim.x + threadIdx.x) >> 5;
    int lane = threadIdx.x & 31;
    if (wave >= E) return;
    float a = expv[wave] * (1.0f / (*sumf));
    if (lane == 0) attn_out[wave] = a;
    long long s = src[wave], d = dst[wave];
    unsafeAtomicAdd(&h_agg[d * DIM + lane],      h[s * DIM + lane]      * a);
    unsafeAtomicAdd(&h_agg[d * DIM + 32 + lane], h[s * DIM + 32 + lane] * a);
}

// ---------- 5: h_new = relu(h_agg @ W_t + b_t) + h via V_WMMA_F32_16X16X4_F32 ----------
// One wave owns one 16-row M-tile and computes ALL FOUR 16-col N-tiles with 4
// accumulators, so each A (h_agg) fragment is loaded once and fed to 4 WMMAs.
// 64 WMMAs per wave. Wave-uniform tail guard keeps EXEC all-ones inside WMMA.
__global__ void gemm_wmma_kernel(const float* __restrict__ A,      // h_agg [Nn,64]
                                 const float* __restrict__ W,      // W_t [64,64]
                                 const float* __restrict__ bias,   // b_t [64]
                                 const float* __restrict__ hres,   // h [Nn,64]
                                 float* __restrict__ out, int Nn) {
    const int wave  = threadIdx.x >> 5;
    const int lane  = threadIdx.x & 31;
    const int half  = lane >> 4;          // 0: lanes 0-15, 1: lanes 16-31
    const int l16   = lane & 15;
    const int mtile = blockIdx.x * (blockDim.x >> 5) + wave;
    if (mtile >= (Nn >> 4)) return;       // uniform across the wave

    const long long arow = (long long)mtile * 16 + l16;  // A row (M = lane%16)

    v8f acc0 = {}, acc1 = {}, acc2 = {}, acc3 = {};
    #pragma unroll
    for (int k = 0; k < DIM; k += 4) {
        const int ka = k + 2 * half;      // A: half0 -> K{k,k+1}, half1 -> K{k+2,k+3}
        const v2f a = *(const v2f*)(A + arow * DIM + ka);   // one b64 load, reused 4x
        v2f b0, b1, b2, b3;               // B: K pair striped over N lanes, 4 N-tiles
        b0.x = W[ka * DIM +  0 + l16];  b0.y = W[(ka + 1) * DIM +  0 + l16];
        b1.x = W[ka * DIM + 16 + l16];  b1.y = W[(ka + 1) * DIM + 16 + l16];
        b2.x = W[ka * DIM + 32 + l16];  b2.y = W[(ka + 1) * DIM + 32 + l16];
        b3.x = W[ka * DIM + 48 + l16];  b3.y = W[(ka + 1) * DIM + 48 + l16];
        acc0 = __builtin_amdgcn_wmma_f32_16x16x4_f32(false, a, false, b0, (short)0, acc0, false, false);
        acc1 = __builtin_amdgcn_wmma_f32_16x16x4_f32(false, a, false, b1, (short)0, acc1, false, false);
        acc2 = __builtin_amdgcn_wmma_f32_16x16x4_f32(false, a, false, b2, (short)0, acc2, false, false);
        acc3 = __builtin_amdgcn_wmma_f32_16x16x4_f32(false, a, false, b3, (short)0, acc3, false, false);
    }

    // epilogue: bias + ReLU + residual, per N-tile
    const float bn0 = bias[ 0 + l16];
    const float bn1 = bias[16 + l16];
    const float bn2 = bias[32 + l16];
    const float bn3 = bias[48 + l16];
    #pragma unroll
    for (int r = 0; r < 8; ++r) {         // C/D: VGPR r holds rows {r, r+8}
        const long long m = (long long)mtile * 16 + r + 8 * half;
        const long long base = m * DIM + l16;
        float v0 = acc0[r] + bn0; v0 = v0 > 0.0f ? v0 : 0.0f;
        float v1 = acc1[r] + bn1; v1 = v1 > 0.0f ? v1 : 0.0f;
        float v2 = acc2[r] + bn2; v2 = v2 > 0.0f ? v2 : 0.0f;
        float v3 = acc3[r] + bn3; v3 = v3 > 0.0f ? v3 : 0.0f;
        out[base +  0] = v0 + hres[base +  0];
        out[base + 16] = v1 + hres[base + 16];
        out[base + 32] = v2 + hres[base + 32];
        out[base + 48] = v3 + hres[base + 48];
    }
}

extern "C" void kernel_launch(void* const* d_in, const int* in_sizes, int n_in,
                              void* d_out, int out_size, void* d_ws, size_t ws_size,
                              hipStream_t stream) {
    const float* h     = (const float*)d_in[0];
    const int*   src   = (const int*)d_in[1];
    const int*   dst   = (const int*)d_in[2];
    const float* W_att = (const float*)d_in[3];
    const float* b_att = (const float*)d_in[4];
    const float* W_t   = (const float*)d_in[5];
    const float* b_t   = (const float*)d_in[6];

    const int E  = in_sizes[1];
    const int Nn = in_sizes[0] / DIM;

    // workspace layout: [logits/exp E floats][h_agg Nn*64 floats][maxkey u32][sum f32]
    float*    logits = (float*)d_ws;
    float*    h_agg  = logits + E;
    unsigned* maxkey = (unsigned*)(h_agg + (size_t)Nn * DIM);
    float*    sumf   = (float*)(maxkey + 1);

    float* out_h    = (float*)d_out;                 // [Nn,64]
    float* out_attn = out_h + (size_t)Nn * DIM;      // [E]

    // 0: zero h_agg + stats (maxkey=0 and sum=0.0f share the zeroed region)
    const int zn = Nn * DIM + 2;
    zero_kernel<<<(zn + 255) / 256, 256, 0, stream>>>(h_agg, zn);

    // wave-per-edge kernels: 256 threads = 8 waves/block (wave32)
    const int eblocks = (E + 7) / 8;
    edge_logits_kernel<<<eblocks, 256, 0, stream>>>(h, src, dst, W_att, b_att, logits, E);

    max_kernel<<<512, 256, 0, stream>>>(logits, E, maxkey);
    expsum_kernel<<<512, 256, 0, stream>>>(logits, E, maxkey, sumf);

    scatter_kernel<<<eblocks, 256, 0, stream>>>(h, src, dst, logits, sumf,
                                                out_attn, h_agg, E);

    // WMMA GEMM: 8 waves/block, each wave = one M-tile x all 4 N-tiles
    const int nMtiles = Nn / 16;                     // 3125
    gemm_wmma_kernel<<<(nMtiles + 7) / 8, 256, 0, stream>>>(h_agg, W_t, b_t, h,
                                                            out_h, Nn);
}